// manifoldNet_2396591751488
// MI455X (gfx1250) — compile-verified
//
#include <hip/hip_runtime.h>
#include <cstddef>

#define BLOCK_K1 128
#define NSWEEP1  8
#define NSWEEP2  8
#define REEIG_EPS 1.0e-4f

#define HSTRIDE 124   // H rows padded 121 -> 124 (multiple of 4, b64-aligned)
#define WLSTRIDE 126  // LDS Wl tile K-stride (even -> 8B aligned; 126%64=62 -> no bank conflicts)

typedef __attribute__((ext_vector_type(2))) float v2f;
typedef __attribute__((ext_vector_type(8))) float v8f;

// ---------------------------------------------------------------------------
// Per-thread cyclic Jacobi eigensolver on an n x n (n<=12) symmetric matrix.
// A and U live in LDS with [element][thread] layout: elem e of this thread is
// at ptr[e*BLOCK_K1 + tid] (stride-1 across lanes -> bank-conflict free).
// On exit: diag(A) = eigenvalues, U = eigenvectors (columns).
// ---------------------------------------------------------------------------
__device__ __forceinline__ void jacobi_lds(float* __restrict__ A,
                                           float* __restrict__ U,
                                           int tid, int n, int sweeps) {
  // U = I (full 12x12 region; extra row/col harmless when n == 11)
  for (int e = 0; e < 144; ++e)
    U[e * BLOCK_K1 + tid] = (e % 13 == 0) ? 1.0f : 0.0f;

  for (int s = 0; s < sweeps; ++s) {
    for (int p = 0; p < n - 1; ++p) {
      for (int q = p + 1; q < n; ++q) {
        float apq = A[(p * 12 + q) * BLOCK_K1 + tid];
        float app = A[(p * 12 + p) * BLOCK_K1 + tid];
        float aqq = A[(q * 12 + q) * BLOCK_K1 + tid];
        // Branch-free rotation; theta=inf/NaN path is overridden below.
        float theta = (aqq - app) / (2.0f * apq);
        float t = __builtin_copysignf(1.0f, theta) /
                  (fabsf(theta) + sqrtf(1.0f + theta * theta));
        float c = rsqrtf(1.0f + t * t);
        float sn = t * c;
        if (!(fabsf(apq) > 1.0e-12f)) { c = 1.0f; sn = 0.0f; }
        // A := A * G   (columns p,q)
        for (int k = 0; k < n; ++k) {
          float akp = A[(k * 12 + p) * BLOCK_K1 + tid];
          float akq = A[(k * 12 + q) * BLOCK_K1 + tid];
          A[(k * 12 + p) * BLOCK_K1 + tid] = c * akp - sn * akq;
          A[(k * 12 + q) * BLOCK_K1 + tid] = sn * akp + c * akq;
        }
        // A := G^T * A (rows p,q)
        for (int k = 0; k < n; ++k) {
          float apk = A[(p * 12 + k) * BLOCK_K1 + tid];
          float aqk = A[(q * 12 + k) * BLOCK_K1 + tid];
          A[(p * 12 + k) * BLOCK_K1 + tid] = c * apk - sn * aqk;
          A[(q * 12 + k) * BLOCK_K1 + tid] = sn * apk + c * aqk;
        }
        // U := U * G
        for (int k = 0; k < n; ++k) {
          float ukp = U[(k * 12 + p) * BLOCK_K1 + tid];
          float ukq = U[(k * 12 + q) * BLOCK_K1 + tid];
          U[(k * 12 + p) * BLOCK_K1 + tid] = c * ukp - sn * ukq;
          U[(k * 12 + q) * BLOCK_K1 + tid] = sn * ukp + c * ukq;
        }
      }
    }
  }
}

// ---------------------------------------------------------------------------
// Fused SPDNet pipeline: bimap1 -> eigh -> (fold reeig+bimap2) -> eigh ->
// log-reconstruct. WRITE_H: emit H (B x HSTRIDE, zero-padded) to ws for the
// WMMA GEMM; otherwise fuse the 121->52 linear layer per-thread.
// ---------------------------------------------------------------------------
template <bool WRITE_H>
__global__ __launch_bounds__(BLOCK_K1) void spdnet_kernel(
    const float* __restrict__ x, const float* __restrict__ w1,
    const float* __restrict__ w2, const float* __restrict__ wl,
    const float* __restrict__ bl, float* __restrict__ outOrH, int B) {
  extern __shared__ float lds[];
  const int tid  = threadIdx.x;
  const int bidx = blockIdx.x * BLOCK_K1 + tid;
  const bool valid = bidx < B;

  float* sW1 = lds;            // 144 floats, shared (broadcast reads)
  float* sW2 = lds + 144;      // 132 floats
  float* A   = lds + 304;                    // per-thread 12x12
  float* U   = lds + 304 + 144 * BLOCK_K1;   // per-thread 12x12

  if (tid < 144) sW1[tid] = w1[tid];
  if (tid < 132) sW2[tid] = w2[tid];
  __syncthreads();

  // Load this thread's X (144 floats) into its LDS column (b128 loads).
  if (valid) {
    const float4* xv = (const float4*)(x + (size_t)bidx * 144);
#pragma unroll
    for (int e4 = 0; e4 < 36; ++e4) {
      float4 v = xv[e4];
      A[(e4 * 4 + 0) * BLOCK_K1 + tid] = v.x;
      A[(e4 * 4 + 1) * BLOCK_K1 + tid] = v.y;
      A[(e4 * 4 + 2) * BLOCK_K1 + tid] = v.z;
      A[(e4 * 4 + 3) * BLOCK_K1 + tid] = v.w;
    }
  }

  // bimap1: T = X*W1 (into U region), then A = W1^T * T.
  for (int i = 0; i < 12; ++i) {
    float xr[12];
#pragma unroll
    for (int j = 0; j < 12; ++j) xr[j] = A[(i * 12 + j) * BLOCK_K1 + tid];
    for (int l = 0; l < 12; ++l) {
      float acc = 0.0f;
#pragma unroll
      for (int j = 0; j < 12; ++j) acc += xr[j] * sW1[j * 12 + l];
      U[(i * 12 + l) * BLOCK_K1 + tid] = acc;
    }
  }
  for (int l = 0; l < 12; ++l) {
    float tc[12];
#pragma unroll
    for (int i = 0; i < 12; ++i) tc[i] = U[(i * 12 + l) * BLOCK_K1 + tid];
    for (int k = 0; k < 12; ++k) {
      float acc = 0.0f;
#pragma unroll
      for (int i = 0; i < 12; ++i) acc += sW1[i * 12 + k] * tc[i];
      A[(k * 12 + l) * BLOCK_K1 + tid] = acc;
    }
  }

  // Eigensolve #1 (12x12).
  jacobi_lds(A, U, tid, 12, NSWEEP1);

  // Fold reeig + bimap2: Vs[k][l] = sqrt(max(w_k,eps)) * sum_i U[i][k]*W2[i][l]
  // stored transposed into the U region at U[l*12+k].
  for (int k = 0; k < 12; ++k) {
    float u[12];
#pragma unroll
    for (int i = 0; i < 12; ++i) u[i] = U[(i * 12 + k) * BLOCK_K1 + tid];
    float wk = A[(k * 12 + k) * BLOCK_K1 + tid];
    float sw = sqrtf(fmaxf(wk, REEIG_EPS));
    for (int l = 0; l < 11; ++l) {
      float acc = 0.0f;
#pragma unroll
      for (int i = 0; i < 12; ++i) acc += u[i] * sW2[i * 11 + l];
      U[(l * 12 + k) * BLOCK_K1 + tid] = sw * acc;
    }
  }
  // A2[a][b] = sum_k Vs[k][a]*Vs[k][b]  (Gram of Vs rows) -> A region.
  for (int a = 0; a < 11; ++a) {
    float ra[12];
#pragma unroll
    for (int k = 0; k < 12; ++k) ra[k] = U[(a * 12 + k) * BLOCK_K1 + tid];
    for (int b = 0; b < 11; ++b) {
      float acc = 0.0f;
#pragma unroll
      for (int k = 0; k < 12; ++k) acc += ra[k] * U[(b * 12 + k) * BLOCK_K1 + tid];
      A[(a * 12 + b) * BLOCK_K1 + tid] = acc;
    }
  }

  // Eigensolve #2 (11x11). logeig(reeig(.)) shares these eigenvectors.
  jacobi_lds(A, U, tid, 11, NSWEEP2);

  // Column scale: A[:,k] = log(max(w2_k,eps)) * U[:,k].
  for (int k = 0; k < 11; ++k) {
    float lw = logf(fmaxf(A[(k * 12 + k) * BLOCK_K1 + tid], REEIG_EPS));
    for (int a = 0; a < 11; ++a)
      A[(a * 12 + k) * BLOCK_K1 + tid] = lw * U[(a * 12 + k) * BLOCK_K1 + tid];
  }

  // H[a][b] = sum_k A[a][k] * U[b][k]
  if (!valid) return;
  if (WRITE_H) {
    for (int a = 0; a < 11; ++a) {
      float ra[11];
#pragma unroll
      for (int k = 0; k < 11; ++k) ra[k] = A[(a * 12 + k) * BLOCK_K1 + tid];
      for (int b = 0; b < 11; ++b) {
        float acc = 0.0f;
#pragma unroll
        for (int k = 0; k < 11; ++k) acc += ra[k] * U[(b * 12 + k) * BLOCK_K1 + tid];
        outOrH[(size_t)bidx * HSTRIDE + a * 11 + b] = acc;
      }
    }
    // Zero the K padding so the GEMM needs no tail conditionals.
    outOrH[(size_t)bidx * HSTRIDE + 121] = 0.0f;
    outOrH[(size_t)bidx * HSTRIDE + 122] = 0.0f;
    outOrH[(size_t)bidx * HSTRIDE + 123] = 0.0f;
  } else {
    float acc[52];
#pragma unroll
    for (int c = 0; c < 52; ++c) acc[c] = bl[c];
    for (int a = 0; a < 11; ++a) {
      float ra[11];
#pragma unroll
      for (int k = 0; k < 11; ++k) ra[k] = A[(a * 12 + k) * BLOCK_K1 + tid];
      for (int b = 0; b < 11; ++b) {
        float h = 0.0f;
#pragma unroll
        for (int k = 0; k < 11; ++k) h += ra[k] * U[(b * 12 + k) * BLOCK_K1 + tid];
        int f = a * 11 + b;
#pragma unroll
        for (int c = 0; c < 52; ++c) acc[c] += h * wl[c * 121 + f];  // uniform
      }
    }
#pragma unroll
    for (int c = 0; c < 52; ++c) outOrH[(size_t)bidx * 52 + c] = acc[c];
  }
}

// ---------------------------------------------------------------------------
// out[B,52] = H[B,121(pad124)] * Wl[52,121]^T + bl via v_wmma_f32_16x16x4_f32.
// Wl is staged once per block into a zero-padded LDS tile (64 x WLSTRIDE), so
// the K-loop has NO conditionals: 1 global b64 + 1 ds b64 + 1 wmma per chunk.
// One wave computes one 16x16 (M x N) tile. EXEC stays all-ones per wave.
// ---------------------------------------------------------------------------
__global__ __launch_bounds__(256) void linear_wmma(
    const float* __restrict__ H, const float* __restrict__ Wl,
    const float* __restrict__ bl, float* __restrict__ out, int B) {
  __shared__ float sWl[64 * WLSTRIDE];

  // Cooperative zero-padded staging of Wl^T tile (all 256 threads, pre-guard).
  for (int i = threadIdx.x; i < 64 * WLSTRIDE; i += 256) {
    int c = i / WLSTRIDE;
    int kk = i - c * WLSTRIDE;
    sWl[i] = (c < 52 && kk < 121) ? Wl[c * 121 + kk] : 0.0f;
  }
  __syncthreads();

  const int lane  = threadIdx.x & 31;
  const int gwave = (blockIdx.x * 256 + threadIdx.x) >> 5;
  const int mt = gwave >> 2;        // 4 N-tiles (52 -> 64)
  const int nt = gwave & 3;
  const int mtiles = (B + 15) >> 4;
  if (mt >= mtiles) return;         // uniform per wave; no later barriers

  const int half = lane >> 4;       // 0: K={k0,k0+1}, 1: K={k0+2,k0+3}
  const int l16  = lane & 15;
  const int row  = mt * 16 + l16;   // A: M = lane%16
  const int col  = nt * 16 + l16;   // B: N = lane%16 (padded cols read zeros)

  const float* hrow = H + (size_t)row * HSTRIDE + 2 * half;
  const float* wcol = sWl + col * WLSTRIDE + 2 * half;

  v8f acc = {};
#pragma unroll
  for (int k0 = 0; k0 < 124; k0 += 4) {   // 31 chunks, zero-padded K tail
    v2f a = *(const v2f*)(hrow + k0);     // 8B-aligned: HSTRIDE even, k even
    v2f b = *(const v2f*)(wcol + k0);     // 8B-aligned: WLSTRIDE even
    acc = __builtin_amdgcn_wmma_f32_16x16x4_f32(
        /*neg_a=*/false, a, /*neg_b=*/false, b,
        /*c_mod=*/(short)0, acc, /*reuse_a=*/false, /*reuse_b=*/false);
  }

  // C/D layout: VGPR r, lane L -> M = r + 8*(L/16), N = L%16.
  if (col < 52) {
    const int mbase = mt * 16 + 8 * half;
    const float bias = bl[col];
#pragma unroll
    for (int r = 0; r < 8; ++r) {
      int m = mbase + r;
      if (m < B) out[(size_t)m * 52 + col] = acc[r] + bias;
    }
  }
}

// ---------------------------------------------------------------------------
extern "C" void kernel_launch(void* const* d_in, const int* in_sizes, int n_in,
                              void* d_out, int out_size, void* d_ws,
                              size_t ws_size, hipStream_t stream) {
  const float* x  = (const float*)d_in[0];
  const float* w1 = (const float*)d_in[1];
  const float* w2 = (const float*)d_in[2];
  const float* wl = (const float*)d_in[3];
  const float* bl = (const float*)d_in[4];
  float* out = (float*)d_out;

  const int B = in_sizes[0] / 144;
  const size_t shmem = (size_t)(304 + 288 * BLOCK_K1) * sizeof(float);
  const int grid1 = (B + BLOCK_K1 - 1) / BLOCK_K1;
  const size_t needH = (size_t)B * HSTRIDE * sizeof(float);

  if (ws_size >= needH) {
    float* Hbuf = (float*)d_ws;
    spdnet_kernel<true><<<grid1, BLOCK_K1, shmem, stream>>>(x, w1, w2, wl, bl,
                                                            Hbuf, B);
    const int mtiles = (B + 15) / 16;
    const int waves  = mtiles * 4;
    const int grid2  = (waves * 32 + 255) / 256;
    linear_wmma<<<grid2, 256, 0, stream>>>(Hbuf, wl, bl, out, B);
  } else {
    spdnet_kernel<false><<<grid1, BLOCK_K1, shmem, stream>>>(x, w1, w2, wl, bl,
                                                             out, B);
  }
}